// SageConv_69398081569112
// MI455X (gfx1250) — compile-verified
//
#include <hip/hip_runtime.h>

typedef __attribute__((ext_vector_type(2))) float v2f;
typedef __attribute__((ext_vector_type(4))) float v4f;
typedef __attribute__((ext_vector_type(8))) float v8f;

#define N_NODES 50000
#define N_EDGES 800000
#define IN_FEAT 128
#define OUT_FEAT 128
#define K_TOT 256          // 2 * IN_FEAT
#define LDS_STRIDE 260     // 256 + 4 pad -> conflict-free ds_load_b64

// ---------------------------------------------------------------------------
// Kernel 0: zero the aggregation + degree workspace (harness poisons ws)
// ---------------------------------------------------------------------------
__global__ void zero_ws_kernel(float* __restrict__ ws, int n) {
    int i = blockIdx.x * blockDim.x + threadIdx.x;
    int stride = gridDim.x * blockDim.x;
    for (; i < n; i += stride) ws[i] = 0.0f;
}

// ---------------------------------------------------------------------------
// Kernel 1: per-edge scatter.  128 threads per edge (2 edges / 256-thr block).
// Coalesced 512B gather of h[src], hardware f32 atomic adds into agg[dst],
// lane f==0 increments degree.  All traffic stays in L2 (51 MB working set).
// ---------------------------------------------------------------------------
__global__ __launch_bounds__(256) void scatter_edges_kernel(
        const float* __restrict__ h,
        const int*   __restrict__ src,
        const int*   __restrict__ dst,
        float*       __restrict__ agg,
        float*       __restrict__ deg) {
    int e = blockIdx.x * 2 + (threadIdx.x >> 7);
    int f = threadIdx.x & 127;
    if (e >= N_EDGES) return;
    int s = src[e];
    int d = dst[e];
    float v = h[(size_t)s * IN_FEAT + f];
    unsafeAtomicAdd(&agg[(size_t)d * IN_FEAT + f], v);   // global_atomic_add_f32
    if (f == 0) unsafeAtomicAdd(&deg[d], 1.0f);
}

// ---------------------------------------------------------------------------
// Kernel 2: out[50000,128] = concat(h, agg/max(deg,1)) @ W^T + bias
// One wave per 16x16 tile, V_WMMA_F32_16X16X4_F32 accumulation over K=256.
// Block = 256 threads = 8 waves = all 8 N-tiles of 16 shared rows.
// ---------------------------------------------------------------------------
__global__ __launch_bounds__(256) void sage_gemm_kernel(
        const float* __restrict__ h,
        const float* __restrict__ W,      // [OUT_FEAT][K_TOT], row-major
        const float* __restrict__ bias,   // [OUT_FEAT]
        const float* __restrict__ agg,
        const float* __restrict__ deg,
        float*       __restrict__ out) {
    __shared__ float Atile[16 * LDS_STRIDE];

    const int mtile = blockIdx.x;
    const int row0  = mtile * 16;
    const int t     = threadIdx.x;

    // ---- Stage A tile (16 rows x 256 cols) into LDS; fold mean-divide ----
    {
        const int ml   = t >> 4;           // local row 0..15
        const int cseg = (t & 15) * 16;    // 16-col segment
        const int m    = row0 + ml;
        const float* srcp;
        float scale = 1.0f;
        if (cseg < IN_FEAT) {
            srcp = h + (size_t)m * IN_FEAT + cseg;          // self features
        } else {
            srcp = agg + (size_t)m * IN_FEAT + (cseg - IN_FEAT);  // neighbor mean
            scale = 1.0f / fmaxf(deg[m], 1.0f);
        }
#pragma unroll
        for (int j = 0; j < 16; j += 4) {
            v4f val = *(const v4f*)(srcp + j);
            val *= scale;
            *(v4f*)(&Atile[ml * LDS_STRIDE + cseg + j]) = val;
        }
    }
    __syncthreads();

    // ---- WMMA K-loop ----
    const int wave = t >> 5;        // 0..7 -> N tile
    const int lane = t & 31;
    const int half = lane >> 4;     // 0: K=k0..k0+1 | 1: K=k0+2..k0+3
    const int nl   = lane & 15;
    const int gn   = wave * 16 + nl;                    // global out column
    const float* Wrow = W + (size_t)gn * K_TOT;         // B[k][gn] = W[gn][k]
    const float* Alds = &Atile[nl * LDS_STRIDE + half * 2];

    v8f acc = {};
#pragma unroll 8
    for (int k0 = 0; k0 < K_TOT; k0 += 4) {
        v2f a = *(const v2f*)(Alds + k0);                 // ds_load_b64
        v2f b = *(const v2f*)(Wrow + k0 + half * 2);      // global_load_b64
        acc = __builtin_amdgcn_wmma_f32_16x16x4_f32(
                /*neg_a=*/false, a, /*neg_b=*/false, b,
                /*c_mod=*/(short)0, acc,
                /*reuse_a=*/false, /*reuse_b=*/false);
    }

    // ---- Epilogue: bias + store (VGPR r = row r / r+8 per half) ----
    const float bn = bias[gn];
#pragma unroll
    for (int r = 0; r < 8; ++r) {
        int row = row0 + half * 8 + r;
        out[(size_t)row * OUT_FEAT + gn] = acc[r] + bn;
    }
}

// ---------------------------------------------------------------------------
extern "C" void kernel_launch(void* const* d_in, const int* in_sizes, int n_in,
                              void* d_out, int out_size, void* d_ws, size_t ws_size,
                              hipStream_t stream) {
    const float* h    = (const float*)d_in[0];
    const int*   src  = (const int*)  d_in[1];
    const int*   dst  = (const int*)  d_in[2];
    const float* W    = (const float*)d_in[3];
    const float* bias = (const float*)d_in[4];
    float*       out  = (float*)d_out;

    // workspace layout: agg [N_NODES*IN_FEAT] then deg [N_NODES]
    float* agg = (float*)d_ws;
    float* deg = agg + (size_t)N_NODES * IN_FEAT;

    const int ztot = N_NODES * IN_FEAT + N_NODES;
    zero_ws_kernel<<<4096, 256, 0, stream>>>(agg, ztot);

    scatter_edges_kernel<<<(N_EDGES + 1) / 2, 256, 0, stream>>>(h, src, dst, agg, deg);

    sage_gemm_kernel<<<N_NODES / 16, 256, 0, stream>>>(h, W, bias, agg, deg, out);
}